// GCN_LSTM_30081950941779
// MI455X (gfx1250) — compile-verified
//
#include <hip/hip_runtime.h>
#include <hip/hip_bf16.h>
#include <math.h>

// ---------------------------------------------------------------------------
// GCN(3 layers, symmetric-norm w/ self loops) + BN + ReLU, then 3-layer LSTM.
// Dense math on the CDNA5 matrix core via V_WMMA_F32_16X16X4_F32 (fp32 in/out,
// matches the fp32 reference). Each wave32 computes a 64x16 output strip
// (4 accumulator tiles) so the B (weight) fragment is loaded once per k-step
// and reused 4x -> 4 WMMAs per ~6 loads instead of 4 per 12. Ragged M is
// handled by clamping A-row pointers (EXEC stays all-ones, required by WMMA)
// and skipping stores for out-of-range tiles (wave-uniform branch).
// ---------------------------------------------------------------------------

namespace {
constexpr int T_  = 8;
constexpr int NN  = 10000;
constexpr int EE  = 200000;
constexpr int FIN = 32;
constexpr int HD  = 128;
constexpr int OD  = 64;
constexpr int G4  = 4 * OD;      // 256 gate columns
constexpr int MU  = 4;           // M-tiles per wave (64 rows)
constexpr float BN_EPS = 1e-5f;
}

typedef float v2f __attribute__((ext_vector_type(2)));
typedef float v8f __attribute__((ext_vector_type(8)));

__device__ __forceinline__ float sigmoidf_(float x) { return 1.0f / (1.0f + __expf(-x)); }

// ---------------- elementwise / graph-prep kernels -------------------------

__global__ void k_zero(float* __restrict__ p, long long n) {
  long long i = (long long)blockIdx.x * blockDim.x + threadIdx.x;
  if (i < n) p[i] = 0.0f;
}

__global__ void k_degree(const int* __restrict__ dst, float* __restrict__ deg) {
  int e = blockIdx.x * blockDim.x + threadIdx.x;
  if (e < EE) atomicAdd(&deg[dst[e]], 1.0f);
}

__global__ void k_dis(float* __restrict__ deg_dis, float* __restrict__ self_w) {
  int n = blockIdx.x * blockDim.x + threadIdx.x;
  if (n < NN) {
    float d = rsqrtf(deg_dis[n] + 1.0f);   // deg includes self-loop (+1)
    deg_dis[n] = d;
    self_w[n]  = d * d;
  }
}

__global__ void k_coef(const int* __restrict__ src, const int* __restrict__ dst,
                       const float* __restrict__ dis, float* __restrict__ coef) {
  int e = blockIdx.x * blockDim.x + threadIdx.x;
  if (e < EE) coef[e] = dis[src[e]] * dis[dst[e]];
}

// ---------------- WMMA fp32 GEMM: C[M,O] = A[M,K] * B[K,O] -----------------
// A-fragment layout per CDNA5 ISA 7.12.2 (16x4 f32):
//   lanes 0-15  : VGPR0=K(k0+0), VGPR1=K(k0+1), row M = lane
//   lanes 16-31 : VGPR0=K(k0+2), VGPR1=K(k0+3), row M = lane-16
// B/C: rows striped across lanes within a VGPR (col = lane&15).
__global__ void k_gemm_nn(const float* __restrict__ A, const float* __restrict__ B,
                          float* __restrict__ C, int M, int K, int O) {
  const int lane = threadIdx.x;                 // 0..31
  const int m    = lane & 15;
  const int kh   = (lane >> 4) << 1;            // 0 or 2
  const int col  = blockIdx.y * 16 + m;
  const int rb0  = blockIdx.x * (16 * MU);

  const float* arow[MU];
#pragma unroll
  for (int i = 0; i < MU; ++i) {
    int r = rb0 + i * 16 + m;
    r = r < M ? r : M - 1;                      // clamp: keep EXEC all-ones
    arow[i] = A + (size_t)r * K;
    __builtin_prefetch(arow[i], 0, 3);          // global_prefetch_b8
  }

  v8f acc[MU];
#pragma unroll
  for (int i = 0; i < MU; ++i) acc[i] = v8f{0.f,0.f,0.f,0.f,0.f,0.f,0.f,0.f};

  for (int k0 = 0; k0 < K; k0 += 4) {
    v2f b;                                      // B fragment shared by MU tiles
    b.x = B[(size_t)(k0 + kh)     * O + col];
    b.y = B[(size_t)(k0 + kh + 1) * O + col];
#pragma unroll
    for (int i = 0; i < MU; ++i) {
      v2f a;
      a.x = arow[i][k0 + kh];
      a.y = arow[i][k0 + kh + 1];
      acc[i] = __builtin_amdgcn_wmma_f32_16x16x4_f32(false, a, false, b,
                                                     (short)0, acc[i], false, false);
    }
  }

  const int vhalf = (lane >> 4) << 3;           // rows v / v+8 per C layout
#pragma unroll
  for (int i = 0; i < MU; ++i) {
    const int r0 = rb0 + i * 16;
    if (r0 < M) {                               // wave-uniform guard
#pragma unroll
      for (int v = 0; v < 8; ++v)
        C[(size_t)(r0 + vhalf + v) * O + col] = acc[i][v];
    }
  }
}

// out[t,n,o] = hw[t,n,o]*self_w[n] + bias[o]   (self-loop term + bias)
__global__ void k_init_out(const float* __restrict__ hw, const float* __restrict__ self_w,
                           const float* __restrict__ bias, float* __restrict__ out) {
  long long i = (long long)blockIdx.x * blockDim.x + threadIdx.x;
  const long long total = (long long)T_ * NN * HD;
  if (i < total) {
    int o = (int)(i % HD);
    int n = (int)((i / HD) % NN);
    out[i] = hw[i] * self_w[n] + bias[o];
  }
}

// out[t,dst,:] += hw[t,src,:] * coef[e]   (edge scatter, fp32 L2 atomics)
__global__ void k_scatter(const float* __restrict__ hw, const int* __restrict__ src,
                          const int* __restrict__ dst, const float* __restrict__ coef,
                          float* __restrict__ out) {
  const int Q = HD / 4;
  long long idx = (long long)blockIdx.x * blockDim.x + threadIdx.x;
  const long long total = (long long)T_ * EE * Q;
  if (idx >= total) return;
  int o4 = (int)(idx % Q);
  long long te = idx / Q;
  int e = (int)(te % EE);
  int t = (int)(te / EE);
  int s = src[e], d = dst[e];
  float c = coef[e];
  const float4 v = *(const float4*)&hw[((size_t)t * NN + s) * HD + o4 * 4];
  float* po = &out[((size_t)t * NN + d) * HD + o4 * 4];
  atomicAdd(po + 0, v.x * c);
  atomicAdd(po + 1, v.y * c);
  atomicAdd(po + 2, v.z * c);
  atomicAdd(po + 3, v.w * c);
}

// h = relu(h); optionally eval-mode BatchNorm1d
__global__ void k_bn_relu(float* __restrict__ h, const float* __restrict__ g,
                          const float* __restrict__ b, const float* __restrict__ m,
                          const float* __restrict__ v, int do_bn) {
  long long i = (long long)blockIdx.x * blockDim.x + threadIdx.x;
  const long long total = (long long)T_ * NN * HD;
  if (i >= total) return;
  float x = h[i];
  x = x > 0.0f ? x : 0.0f;
  if (do_bn) {
    int o = (int)(i % HD);
    x = (x - m[o]) * (g[o] * rsqrtf(v[o] + BN_EPS)) + b[o];
  }
  h[i] = x;
}

// ---------------- LSTM gate GEMM (two fused WMMA GEMMs) --------------------
// gates[n,j] = bih[j]+bhh[j] + sum_k X[n,k]*Wih[j,k] + sum_k Hp[n,k]*Whh[j,k]
// Wih is [G4, Kin] row-major (torch layout) => B[k][col] = Wih[col*Kin + k].
// 64x16 strip per wave: weight fragments loaded once per k-step, reused 4x.
__global__ void k_lstm_gates(const float* __restrict__ X, int Kin,
                             const float* __restrict__ Hp,
                             const float* __restrict__ Wih, const float* __restrict__ Whh,
                             const float* __restrict__ bih, const float* __restrict__ bhh,
                             float* __restrict__ G) {
  const int lane = threadIdx.x;
  const int m    = lane & 15;
  const int kh   = (lane >> 4) << 1;
  const int col  = blockIdx.y * 16 + m;         // gate column (O = 256)
  const int rb0  = blockIdx.x * (16 * MU);      // node-row strip

  const float* xrow[MU];
  const float* hrow[MU];
#pragma unroll
  for (int i = 0; i < MU; ++i) {
    int r = rb0 + i * 16 + m;
    r = r < NN ? r : NN - 1;                    // clamp: keep EXEC all-ones
    xrow[i] = X  + (size_t)r * Kin;
    hrow[i] = Hp + (size_t)r * OD;
    __builtin_prefetch(xrow[i], 0, 3);
  }

  v8f acc[MU];
#pragma unroll
  for (int i = 0; i < MU; ++i) acc[i] = v8f{0.f,0.f,0.f,0.f,0.f,0.f,0.f,0.f};

  const float* wi = Wih + (size_t)col * Kin;    // this lane's output column row
  for (int k0 = 0; k0 < Kin; k0 += 4) {
    v2f b;
    b.x = wi[k0 + kh];
    b.y = wi[k0 + kh + 1];
#pragma unroll
    for (int i = 0; i < MU; ++i) {
      v2f a;
      a.x = xrow[i][k0 + kh];
      a.y = xrow[i][k0 + kh + 1];
      acc[i] = __builtin_amdgcn_wmma_f32_16x16x4_f32(false, a, false, b,
                                                     (short)0, acc[i], false, false);
    }
  }
  const float* wh = Whh + (size_t)col * OD;
  for (int k0 = 0; k0 < OD; k0 += 4) {
    v2f b;
    b.x = wh[k0 + kh];
    b.y = wh[k0 + kh + 1];
#pragma unroll
    for (int i = 0; i < MU; ++i) {
      v2f a;
      a.x = hrow[i][k0 + kh];
      a.y = hrow[i][k0 + kh + 1];
      acc[i] = __builtin_amdgcn_wmma_f32_16x16x4_f32(false, a, false, b,
                                                     (short)0, acc[i], false, false);
    }
  }

  const float bias = bih[col] + bhh[col];
  const int vhalf = (lane >> 4) << 3;
#pragma unroll
  for (int i = 0; i < MU; ++i) {
    const int r0 = rb0 + i * 16;
    if (r0 < NN) {                              // wave-uniform guard
#pragma unroll
      for (int v = 0; v < 8; ++v)
        G[(size_t)(r0 + vhalf + v) * G4 + col] = acc[i][v] + bias;
    }
  }
}

// pointwise LSTM cell: gate order (i, f, g, o)
__global__ void k_lstm_cell(const float* __restrict__ G, float* __restrict__ h,
                            float* __restrict__ c, float* __restrict__ yout) {
  int i = blockIdx.x * blockDim.x + threadIdx.x;   // over N*OD
  if (i >= NN * OD) return;
  int n = i / OD, j = i % OD;
  const float* g = G + (size_t)n * G4;
  float ig = sigmoidf_(g[j]);
  float fg = sigmoidf_(g[OD + j]);
  float gg = tanhf(g[2 * OD + j]);
  float og = sigmoidf_(g[3 * OD + j]);
  float cn = fg * c[i] + ig * gg;
  float hn = og * tanhf(cn);
  c[i] = cn;
  h[i] = hn;
  if (yout) yout[i] = hn;
}

// ---------------------------------------------------------------------------

extern "C" void kernel_launch(void* const* d_in, const int* in_sizes, int n_in,
                              void* d_out, int out_size, void* d_ws, size_t ws_size,
                              hipStream_t stream) {
  (void)in_sizes; (void)n_in; (void)out_size; (void)ws_size;

  const float* x       = (const float*)d_in[0];
  const int*   ei      = (const int*)d_in[1];
  const float* w_gcn[3] = {(const float*)d_in[2], (const float*)d_in[4], (const float*)d_in[6]};
  const float* b_gcn[3] = {(const float*)d_in[3], (const float*)d_in[5], (const float*)d_in[7]};
  const float* bn_g[2]  = {(const float*)d_in[8],  (const float*)d_in[12]};
  const float* bn_b[2]  = {(const float*)d_in[9],  (const float*)d_in[13]};
  const float* bn_m[2]  = {(const float*)d_in[10], (const float*)d_in[14]};
  const float* bn_v[2]  = {(const float*)d_in[11], (const float*)d_in[15]};
  const float* w_ih[3]  = {(const float*)d_in[16], (const float*)d_in[20], (const float*)d_in[24]};
  const float* w_hh[3]  = {(const float*)d_in[17], (const float*)d_in[21], (const float*)d_in[25]};
  const float* b_ih[3]  = {(const float*)d_in[18], (const float*)d_in[22], (const float*)d_in[26]};
  const float* b_hh[3]  = {(const float*)d_in[19], (const float*)d_in[23], (const float*)d_in[27]};

  const int* srcv = ei;        // edge_index[0,:]
  const int* dstv = ei + EE;   // edge_index[1,:]

  // ---- workspace layout (floats); required ~124 MB with buffer reuse ----
  float* W      = (float*)d_ws;
  float* dis    = W;                               // N   (deg, then rsqrt(deg))
  float* self_w = W + NN;                          // N
  float* coef   = W + 2 * NN;                      // E
  const size_t BIG = (size_t)T_ * NN * HD;         // 10.24M floats
  float* bufA = coef + EE;                         // hw scratch / LSTM gates+h+c
  float* bufB = bufA + BIG;                        // h1 / h3 (LSTM input seq)
  float* bufC = bufB + BIG;                        // h2 / LSTM ys ping-pong

  // ---- graph normalization ----
  k_zero  <<<(NN + 255) / 256, 256, 0, stream>>>(dis, NN);
  k_degree<<<(EE + 255) / 256, 256, 0, stream>>>(dstv, dis);
  k_dis   <<<(NN + 255) / 256, 256, 0, stream>>>(dis, self_w);
  k_coef  <<<(EE + 255) / 256, 256, 0, stream>>>(srcv, dstv, dis, coef);

  // ---- 3 GCN layers ----
  struct Layer { const float* in; int K; float* hw; float* out; };
  Layer L[3] = { { x,    FIN, bufA, bufB },
                 { bufB, HD,  bufA, bufC },
                 { bufC, HD,  bufA, bufB } };
  const long long tot_h = (long long)T_ * NN * HD;
  const long long tot_s = (long long)T_ * EE * (HD / 4);
  const int M = T_ * NN;                           // 80000, divisible by 64
  for (int l = 0; l < 3; ++l) {
    dim3 g((M + 16 * MU - 1) / (16 * MU), HD / 16);   // (1250, 8) strips
    k_gemm_nn <<<g, 32, 0, stream>>>(L[l].in, w_gcn[l], L[l].hw, M, L[l].K, HD);
    k_init_out<<<(unsigned)((tot_h + 255) / 256), 256, 0, stream>>>(L[l].hw, self_w, b_gcn[l], L[l].out);
    k_scatter <<<(unsigned)((tot_s + 255) / 256), 256, 0, stream>>>(L[l].hw, srcv, dstv, coef, L[l].out);
    if (l < 2)
      k_bn_relu<<<(unsigned)((tot_h + 255) / 256), 256, 0, stream>>>(L[l].out, bn_g[l], bn_b[l], bn_m[l], bn_v[l], 1);
    else
      k_bn_relu<<<(unsigned)((tot_h + 255) / 256), 256, 0, stream>>>(L[l].out, nullptr, nullptr, nullptr, nullptr, 0);
  }
  // bufB now holds h3 [T,N,HD]; bufA, bufC free.

  // ---- 3-layer LSTM (reuse bufA for gates/h/c, bufC for ys ping-pong) ----
  float* gates = bufA;                             // N*256
  float* hb    = bufA + (size_t)NN * G4;           // N*64
  float* cb    = hb   + (size_t)NN * OD;           // N*64
  float* ysA   = bufC;                             // T*N*64
  float* ysB   = bufC + (size_t)T_ * NN * OD;      // T*N*64

  const float* seq_in[3]  = { bufB, ysA, ysB };
  float*       seq_out[3] = { ysA,  ysB, nullptr };
  const int    kin[3]     = { HD, OD, OD };

  for (int l = 0; l < 3; ++l) {
    k_zero<<<(2 * NN * OD + 255) / 256, 256, 0, stream>>>(hb, (long long)2 * NN * OD);
    for (int t = 0; t < T_; ++t) {
      const float* Xt = seq_in[l] + (size_t)t * NN * kin[l];
      dim3 g((NN + 16 * MU - 1) / (16 * MU), G4 / 16);  // (157, 16) strips
      k_lstm_gates<<<g, 32, 0, stream>>>(Xt, kin[l], hb, w_ih[l], w_hh[l],
                                         b_ih[l], b_hh[l], gates);
      float* yout = seq_out[l] ? (seq_out[l] + (size_t)t * NN * OD)
                               : ((t == T_ - 1) ? (float*)d_out : nullptr);
      k_lstm_cell<<<(NN * OD + 255) / 256, 256, 0, stream>>>(gates, hb, cb, yout);
    }
  }
}